// Generator_44538810859851
// MI455X (gfx1250) — compile-verified
//
#include <hip/hip_runtime.h>

typedef float v2f __attribute__((ext_vector_type(2)));
typedef float v8f __attribute__((ext_vector_type(8)));

#define N_SITES 64
#define CHI 64
#define BATCH 32768

#define SA_STRIDE 76   // sAT row stride (floats): 76 mod 64 = 12 -> conflict-free B frags, rows 16B aligned
#define SL_STRIDE 68   // sL  row stride (floats): 68 mod 64 = 4  -> conflict-free A frags / L writes
#define XQ_STRIDE 129  // QR column stride: 129 mod 64 = 1 -> conflict-free
#define QP_STRIDE 129

__device__ __forceinline__ void wait_asynccnt0() {
#if __has_builtin(__builtin_amdgcn_s_wait_asynccnt)
  __builtin_amdgcn_s_wait_asynccnt(0);
#else
  asm volatile("s_wait_asynccnt 0x0" ::: "memory");
#endif
}

// Issue one wave-wide async global->LDS b128 copy. ldsOff = wave-relative LDS byte
// offset (low 32 bits of the generic pointer), goff = per-lane byte offset, gbase = SGPR base.
__device__ __forceinline__ void async_b128(unsigned ldsOff, unsigned goff, const float* gbase) {
  asm volatile("global_load_async_to_lds_b128 %0, %1, %2"
               :: "v"(ldsOff), "v"(goff), "s"(gbase)
               : "memory");
}

// ---------------- Phase 1: isometrize (Householder QR, LAPACK sign convention) ----------------
// One block per site. X = M^T (128x64), stored column-major: Xs[c][j]. Thread c owns column c.
// Writes A^T: AT[site][j][l] = Q[j][l]  (so the sampler reads it coalesced & transposed).
__global__ __launch_bounds__(64) void iso_qr_kernel(const float* __restrict__ W,
                                                    float* __restrict__ AT) {
  __shared__ float Xs[64 * XQ_STRIDE];
  __shared__ float Qs[32 * QP_STRIDE];
  __shared__ float taus[64];
  __shared__ float sred[64];
  __shared__ float sbc[4];
  const int c = threadIdx.x;      // 0..63
  const int site = blockIdx.x;    // 0..63
  const float* Wi = W + (size_t)site * CHI * 2 * CHI;  // [64][128] row-major
  for (int j = 0; j < 128; ++j) Xs[c * XQ_STRIDE + j] = Wi[c * 128 + j];
  __syncthreads();

  for (int k = 0; k < 64; ++k) {
    float part = 0.f;
    for (int j = k + c; j < 128; j += 64) {
      float x = Xs[k * XQ_STRIDE + j];
      part += x * x;
    }
    sred[c] = part;
    __syncthreads();
    if (c == 0) {
      float n2 = 0.f;
      for (int t = 0; t < 64; ++t) n2 += sred[t];
      float alpha = Xs[k * XQ_STRIDE + k];
      float nrm = sqrtf(n2);
      float beta = (alpha >= 0.f) ? -nrm : nrm;       // beta = -copysign(nrm, alpha)
      float den = alpha - beta;
      sbc[0] = beta;
      sbc[1] = (den != 0.f) ? (1.f / den) : 0.f;
      float tau = (beta != 0.f) ? ((beta - alpha) / beta) : 0.f;
      sbc[2] = tau;
      taus[k] = tau;
    }
    __syncthreads();
    float beta = sbc[0], invden = sbc[1], tau = sbc[2];
    for (int j = k + 1 + c; j < 128; j += 64) Xs[k * XQ_STRIDE + j] *= invden;
    if (c == 0) Xs[k * XQ_STRIDE + k] = beta;
    __syncthreads();
    if (c > k) {
      float dot = Xs[c * XQ_STRIDE + k];
      for (int j = k + 1; j < 128; ++j) dot += Xs[k * XQ_STRIDE + j] * Xs[c * XQ_STRIDE + j];
      float w = tau * dot;
      Xs[c * XQ_STRIDE + k] -= w;
      for (int j = k + 1; j < 128; ++j) Xs[c * XQ_STRIDE + j] -= w * Xs[k * XQ_STRIDE + j];
    }
    __syncthreads();
  }

  // Q = H_0 ... H_63 * E (thin), formed in two 32-column panels to fit LDS
  float* dst = AT + (size_t)site * 128 * 64;
  for (int p = 0; p < 2; ++p) {
    if (c < 32) {
      const int col = p * 32 + c;
      for (int j = 0; j < 128; ++j) Qs[c * QP_STRIDE + j] = (j == col) ? 1.f : 0.f;
      for (int k = 63; k >= 0; --k) {
        float tau = taus[k];
        float dot = Qs[c * QP_STRIDE + k];
        for (int j = k + 1; j < 128; ++j) dot += Xs[k * XQ_STRIDE + j] * Qs[c * QP_STRIDE + j];
        float w = tau * dot;
        Qs[c * QP_STRIDE + k] -= w;
        for (int j = k + 1; j < 128; ++j) Qs[c * QP_STRIDE + j] -= w * Xs[k * XQ_STRIDE + j];
      }
      for (int j = 0; j < 128; ++j) dst[j * 64 + col] = Qs[c * QP_STRIDE + j];
    }
    __syncthreads();
  }
}

// ---------------- Phase 2: autoregressive perfect sampler (WMMA f32 16x16x4) ----------------
// Block = 128 threads = 4 waves; each wave owns 16 batch rows; 64 rows/block, 512 blocks.
// A tiles double-buffered in LDS, fetched with async global->LDS copies overlapped with WMMA.
__global__ __launch_bounds__(128) void sample_kernel(const float* __restrict__ AT,
                                                     const float* __restrict__ U,
                                                     float* __restrict__ out) {
  __shared__ float sAT[2][128 * SA_STRIDE];  // A_i^T: sAT[buf][j*STRIDE + l]
  __shared__ float sL[64 * SL_STRIDE];       // L rows (wave-private per 16-row slice)
  const int tid = threadIdx.x;
  const int lane = tid & 31;
  const int wave = tid >> 5;
  const int h = lane >> 4;     // 16-lane half
  const int l15 = lane & 15;
  const int j7 = lane & 7;
  const int rowbase = wave * 16;
  const int blockRow = blockIdx.x * 64;
  const v8f vzero = {0.f, 0.f, 0.f, 0.f, 0.f, 0.f, 0.f, 0.f};
  const size_t SHAT = (size_t)BATCH * N_SITES * 2;

  const int jBase = tid >> 4;          // async-copy dest row base for this thread
  const int lOff = (tid & 15) << 2;    // dest column base (floats)

  // prologue: start async copy of site 0 into buffer 0
  {
    const float* gsrc = AT;
#pragma unroll
    for (int it = 0; it < 16; ++it) {
      int j = jBase + it * 8;
      unsigned ldsOff = (unsigned)(uintptr_t)&sAT[0][j * SA_STRIDE + lOff];
      unsigned goff = (unsigned)((tid + it * 128) * 16);
      async_b128(ldsOff, goff, gsrc);
    }
  }

  // L0 = e_0 for every row
  for (int idx = tid; idx < 64 * CHI; idx += 128) {
    int r = idx >> 6, cc = idx & 63;
    sL[r * SL_STRIDE + cc] = (cc == 0) ? 1.f : 0.f;
  }
  wait_asynccnt0();
  __syncthreads();

  for (int site = 0; site < N_SITES; ++site) {
    const float* cAT = sAT[site & 1];

    // kick off async copy of the next site's tile into the other buffer
    if (site + 1 < N_SITES) {
      const float* gsrc = AT + (size_t)(site + 1) * 8192;
      float* dbuf = sAT[(site + 1) & 1];
#pragma unroll
      for (int it = 0; it < 16; ++it) {
        int j = jBase + it * 8;
        unsigned ldsOff = (unsigned)(uintptr_t)&dbuf[j * SA_STRIDE + lOff];
        unsigned goff = (unsigned)((tid + it * 128) * 16);
        async_b128(ldsOff, goff, gsrc);
      }
    }

    // T[16 x 128] = L[16 x 64] @ A_i[64 x 128] via 8 N-tiles x 16 K-steps
    v8f acc[8];
#pragma unroll
    for (int nt = 0; nt < 8; ++nt) acc[nt] = vzero;

    const float* aRow = &sL[(rowbase + l15) * SL_STRIDE + 2 * h];
#pragma unroll
    for (int ks = 0; ks < 16; ++ks) {
      v2f a = *(const v2f*)(aRow + 4 * ks);       // A frag: lanes0-15 K=4ks+{0,1}; lanes16-31 K=4ks+{2,3}
      const int kb = 4 * ks + 2 * h;
#pragma unroll
      for (int nt = 0; nt < 8; ++nt) {
        v2f b = *(const v2f*)&cAT[(nt * 16 + l15) * SA_STRIDE + kb];  // B frag rows K, cols nt*16+lane
        acc[nt] = __builtin_amdgcn_wmma_f32_16x16x4_f32(false, a, false, b, (short)0,
                                                        acc[nt], false, false);
      }
    }

    // per-row p0/p1: reduce squares across the 16-lane half
    float p0[8], p1[8];
#pragma unroll
    for (int j = 0; j < 8; ++j) {
      float q0 = acc[0][j] * acc[0][j] + acc[1][j] * acc[1][j] +
                 acc[2][j] * acc[2][j] + acc[3][j] * acc[3][j];
      float q1 = acc[4][j] * acc[4][j] + acc[5][j] * acc[5][j] +
                 acc[6][j] * acc[6][j] + acc[7][j] * acc[7][j];
      q0 += __shfl_xor(q0, 1, 32);  q1 += __shfl_xor(q1, 1, 32);
      q0 += __shfl_xor(q0, 2, 32);  q1 += __shfl_xor(q1, 2, 32);
      q0 += __shfl_xor(q0, 4, 32);  q1 += __shfl_xor(q1, 4, 32);
      q0 += __shfl_xor(q0, 8, 32);  q1 += __shfl_xor(q1, 8, 32);
      p0[j] = q0;  p1[j] = q1;
    }

    // branch decision: lane handles row h*8 + j7 (duplicated across each 8-lane group)
    const int myGlobalRow = blockRow + rowbase + h * 8 + j7;
    float uval = U[(size_t)myGlobalRow * N_SITES + site];
    float p0m = p0[0], p1m = p1[0];
#pragma unroll
    for (int j = 1; j < 8; ++j) {
      p0m = (j7 == j) ? p0[j] : p0m;
      p1m = (j7 == j) ? p1[j] : p1m;
    }
    bool take1 = (uval >= p0m);
    unsigned mask = (unsigned)__ballot(take1);   // wave32: bit (16h + j) = decision for row h*8+j

    int sbit[8];
    float rsc[8];
#pragma unroll
    for (int j = 0; j < 8; ++j) {
      int s = (mask >> (16 * h + j)) & 1;
      sbit[j] = s;
      float ps = s ? p1[j] : p0[j];
      rsc[j] = 1.f / sqrtf(ps);
    }

    // outputs: one lane per row (lanes 0-7 rows 0-7, lanes 16-23 rows 8-15)
    if (l15 < 8) {
      float ps = take1 ? p1m : p0m;
      float sF = take1 ? 1.f : 0.f;
      float2 oh = make_float2(sF, 1.f - sF);      // one_hot(1-s): s=1 -> [1,0], s=0 -> [0,1]
      *(float2*)&out[(size_t)myGlobalRow * (N_SITES * 2) + site * 2] = oh;
      out[SHAT + (size_t)myGlobalRow * N_SITES + site] = ps;
    }

    // L_new = T_s / sqrt(p_s), written back to the wave-private L tile
#pragma unroll
    for (int j = 0; j < 8; ++j) {
      int r = rowbase + h * 8 + j;
      float sc = rsc[j];
#pragma unroll
      for (int nt = 0; nt < 4; ++nt) {
        float val = (sbit[j] ? acc[4 + nt][j] : acc[nt][j]) * sc;
        sL[r * SL_STRIDE + nt * 16 + l15] = val;
      }
    }

    wait_asynccnt0();   // next-site tile resident (this wave's async ops done)
    __syncthreads();    // all waves done reading cAT + all async copies visible
  }
}

extern "C" void kernel_launch(void* const* d_in, const int* in_sizes, int n_in,
                              void* d_out, int out_size, void* d_ws, size_t ws_size,
                              hipStream_t stream) {
  const float* W = (const float*)d_in[0];   // [64][64][2][64] f32
  const float* U = (const float*)d_in[1];   // [32768][64] f32
  float* out = (float*)d_out;               // s_hat [32768][64][2] ++ prob [32768][64]
  float* AT = (float*)d_ws;                 // A^T per site: [64][128][64] f32 (2 MB)
  iso_qr_kernel<<<64, 64, 0, stream>>>(W, AT);
  sample_kernel<<<BATCH / 64, 128, 0, stream>>>(AT, U, out);
}